// GraphSAGE_17016660426790
// MI455X (gfx1250) — compile-verified
//
#include <hip/hip_runtime.h>

typedef __attribute__((ext_vector_type(2))) float v2f;
typedef __attribute__((ext_vector_type(8))) float v8f;

#define DFEAT 256   // D_IN == D_HID == 256

// ---------------- utility kernels ----------------

__global__ void zero_kernel(float* __restrict__ p, long n) {
    long i = (long)blockIdx.x * blockDim.x + threadIdx.x;
    long stride = (long)gridDim.x * blockDim.x;
    for (; i < n; i += stride) p[i] = 0.0f;
}

// One block (256 threads) per edge: sums[dst] += h[src]; cnt[dst] += 1
__global__ void scatter_accum_kernel(const float* __restrict__ hsrc,
                                     const int* __restrict__ src,
                                     const int* __restrict__ dst,
                                     float* __restrict__ sums,
                                     float* __restrict__ cnt,
                                     int nE) {
    int e = blockIdx.x;
    if (e >= nE) return;
    int s = src[e];
    int d = dst[e];
    const float* row = hsrc + (size_t)s * DFEAT;
    float*       out = sums + (size_t)d * DFEAT;
    atomicAdd(&out[threadIdx.x], row[threadIdx.x]);
    if (threadIdx.x == 0) atomicAdd(&cnt[d], 1.0f);
}

__global__ void inv_kernel(const float* __restrict__ cnt,
                           float* __restrict__ inv, int n) {
    int i = blockIdx.x * blockDim.x + threadIdx.x;
    if (i < n) inv[i] = 1.0f / fmaxf(cnt[i], 1.0f);
}

// ---------------- fused SAGE GEMM (fp32 WMMA, register-blocked) ----------------
// Out[m][n] = act( inv[m] * sum_k Asum[m][k]*Wl[k][n] + sum_k Htgt[m][k]*Wr[k][n] + bias[n] )
//
// One wave32 computes a 16 x (16*NT) output slab with NT independent f32
// accumulator chains (V_WMMA_F32_16X16X4_F32). The A fragment is reused by
// 2*NT WMMAs per k-step. RAGGED=false compiles a branch-free inner loop
// (EXEC stays all-1s trivially); RAGGED=true predicates only B-loads/stores.
template<int NT, bool RAGGED, bool RELU>
__global__ void sage_gemm_wmma(const float* __restrict__ Asum,  // M x 256 (neighbor sums)
                               const float* __restrict__ invc,  // M     (1/count)
                               const float* __restrict__ Htgt,  // M x 256 (target feats)
                               const float* __restrict__ Wl,    // 256 x Nact (row major)
                               const float* __restrict__ Wr,    // 256 x Nact
                               const float* __restrict__ bias,  // Nact
                               float* __restrict__ Out,         // M x Nact
                               int tilesNG,      // # of (16*NT)-wide column groups
                               int totalTiles,   // Mtiles * tilesNG
                               int Nact) {
    const int wave = threadIdx.x >> 5;
    const int lane = threadIdx.x & 31;
    const int gw = blockIdx.x * (blockDim.x >> 5) + wave;
    if (gw >= totalTiles) return;   // wave-uniform: EXEC all-1s below

    const int tileM  = gw / tilesNG;
    const int tileNG = gw % tilesNG;
    const int l15  = lane & 15;
    const int row  = tileM * 16 + l15;       // A-fragment row for this lane
    const int koff = (lane < 16) ? 0 : 2;    // f32 A/B layout: hi half-wave holds K+2,K+3
    const int colBase = tileNG * (16 * NT);

    int col[NT];
#pragma unroll
    for (int j = 0; j < NT; ++j) col[j] = colBase + j * 16 + l15;

    const float invr = invc[row];

    v8f acc[NT];
#pragma unroll
    for (int j = 0; j < NT; ++j) {
        float bv = (!RAGGED || col[j] < Nact) ? bias[col[j]] : 0.0f;
#pragma unroll
        for (int i = 0; i < 8; ++i) acc[j][i] = bv;   // n depends only on lane
    }

    // strength-reduced pointers (no in-loop index multiplies)
    const float* arow = Asum + (size_t)row * DFEAT + koff;
    const float* hrow = Htgt + (size_t)row * DFEAT + koff;
    const float* pl = Wl + (size_t)koff * Nact;
    const float* pr = Wr + (size_t)koff * Nact;
    const size_t bstep = 4 * (size_t)Nact;

    for (int k = 0; k < DFEAT; k += 4) {
        // A fragments: contiguous pair -> b64 loads
        float2 av = *(const float2*)arow;  arow += 4;
        float2 hv = *(const float2*)hrow;  hrow += 4;
        v2f a;  a.x  = av.x * invr; a.y  = av.y * invr;   // fold mean divide into A
        v2f a2; a2.x = hv.x;        a2.y = hv.y;

#pragma unroll
        for (int j = 0; j < NT; ++j) {
            v2f b, b2;
            if (!RAGGED || col[j] < Nact) {
                b.x  = pl[col[j]];
                b.y  = pl[Nact + col[j]];
                b2.x = pr[col[j]];
                b2.y = pr[Nact + col[j]];
            } else {
                b.x = 0.0f; b.y = 0.0f; b2.x = 0.0f; b2.y = 0.0f;
            }
            acc[j] = __builtin_amdgcn_wmma_f32_16x16x4_f32(
                         false, a, false, b, (short)0, acc[j], false, false);
            acc[j] = __builtin_amdgcn_wmma_f32_16x16x4_f32(
                         false, a2, false, b2, (short)0, acc[j], false, false);
        }
        pl += bstep;
        pr += bstep;
    }

    // C/D layout: vgpr i -> M = tileM*16 + i (+8 for lanes 16-31), N = col[j]
    const int mbase = tileM * 16 + ((lane < 16) ? 0 : 8);
#pragma unroll
    for (int j = 0; j < NT; ++j) {
        if (!RAGGED || col[j] < Nact) {
#pragma unroll
            for (int i = 0; i < 8; ++i) {
                float v = acc[j][i];
                if (RELU) v = fmaxf(v, 0.0f);
                Out[(size_t)(mbase + i) * Nact + col[j]] = v;
            }
        }
    }
}

// ---------------- launch ----------------

extern "C" void kernel_launch(void* const* d_in, const int* in_sizes, int n_in,
                              void* d_out, int out_size, void* d_ws, size_t ws_size,
                              hipStream_t stream) {
    const float* h   = (const float*)d_in[0];
    const float* wl0 = (const float*)d_in[1];
    const float* wr0 = (const float*)d_in[2];
    const float* b0  = (const float*)d_in[3];
    const float* wl1 = (const float*)d_in[4];
    const float* wr1 = (const float*)d_in[5];
    const float* b1  = (const float*)d_in[6];
    const int*   ei0 = (const int*)d_in[7];
    const int*   ei1 = (const int*)d_in[8];

    const int E0 = in_sizes[7] / 2;   // (2, E0) flat
    const int E1 = in_sizes[8] / 2;
    const int NT0 = 20000;            // n_tgt0
    const int NT1 = 2000;             // n_tgt1
    const int NCLS = 47;

    // workspace partition (floats): ~43 MB total
    float* sums0 = (float*)d_ws;                       // NT0*256
    float* cnt0  = sums0 + (size_t)NT0 * DFEAT;        // NT0
    float* inv0  = cnt0 + NT0;                         // NT0
    float* h1    = inv0 + NT0;                         // NT0*256
    float* sums1 = h1 + (size_t)NT0 * DFEAT;           // NT1*256
    float* cnt1  = sums1 + (size_t)NT1 * DFEAT;        // NT1
    float* inv1  = cnt1 + NT1;                         // NT1

    // 1) zero accumulators (deterministic across graph replays)
    zero_kernel<<<2048, 256, 0, stream>>>(sums0, (long)NT0 * DFEAT);
    zero_kernel<<<64,   256, 0, stream>>>(cnt0, NT0);
    zero_kernel<<<512,  256, 0, stream>>>(sums1, (long)NT1 * DFEAT);
    zero_kernel<<<8,    256, 0, stream>>>(cnt1, NT1);

    // 2) layer-0 aggregation (memory-bound: ~307 MB of gathers + atomics)
    scatter_accum_kernel<<<E0, DFEAT, 0, stream>>>(h, ei0, ei0 + E0, sums0, cnt0, E0);
    inv_kernel<<<(NT0 + 255) / 256, 256, 0, stream>>>(cnt0, inv0, NT0);

    // 3) layer-0 fused GEMM + ReLU  (M=20000, N=256, K=256), 16x64 slab / wave
    {
        const int tilesNG = DFEAT / 64;               // 4 column groups
        const int total   = (NT0 / 16) * tilesNG;     // 5000 waves
        const int blocks  = (total + 7) / 8;          // 8 waves / 256-thread block
        sage_gemm_wmma<4, false, true><<<blocks, 256, 0, stream>>>(
            sums0, inv0, h, wl0, wr0, b0, h1, tilesNG, total, DFEAT);
    }

    // 4) layer-1 aggregation over h1
    scatter_accum_kernel<<<E1, DFEAT, 0, stream>>>(h1, ei1, ei1 + E1, sums1, cnt1, E1);
    inv_kernel<<<(NT1 + 255) / 256, 256, 0, stream>>>(cnt1, inv1, NT1);

    // 5) layer-1 GEMM, no activation  (M=2000, N=47 ragged, K=256), 16x48 slab / wave
    {
        const int tilesNG = 1;                        // 48 cols >= 47
        const int total   = (NT1 / 16) * tilesNG;     // 125 waves
        const int blocks  = (total + 7) / 8;
        sage_gemm_wmma<3, true, false><<<blocks, 256, 0, stream>>>(
            sums1, inv1, h1, wl1, wr1, b1, (float*)d_out, tilesNG, total, NCLS);
    }
}